// GATLayer_17858474017377
// MI455X (gfx1250) — compile-verified
//
#include <hip/hip_runtime.h>

#define Nn 4096
#define IN_Fc 256
#define Hh 8
#define Fc 64
#define HF 512
#define NEG_INF (-1e9f)

typedef float v2f __attribute__((ext_vector_type(2)));
typedef float v8f __attribute__((ext_vector_type(8)));
typedef unsigned int v4u __attribute__((ext_vector_type(4)));
typedef int v4i __attribute__((ext_vector_type(4)));
typedef int v8i __attribute__((ext_vector_type(8)));

__device__ __forceinline__ float lrelu(float x) { return x > 0.0f ? x : 0.2f * x; }

// ---------------------------------------------------------------------------
// Kernel 1: Wh = h @ W   [4096,256] x [256,512] -> [4096,512]
// f32 WMMA 16x16x4. Grid (N/16, 512/64), block 128 (4 waves, 16 cols each).
// A layout (ISA 7.12.2): lane<16 -> {K0,K1}, lane>=16 -> {K2,K3}, M=lane&15.
// ---------------------------------------------------------------------------
__global__ __launch_bounds__(128) void k_wh(const float* __restrict__ h,
                                            const float* __restrict__ W,
                                            float* __restrict__ Wh) {
  const int lane = threadIdx.x & 31;
  const int wave = threadIdx.x >> 5;
  const int r    = lane & 15;
  const int hi   = lane >> 4;                 // 0 or 1
  const int i0   = blockIdx.x * 16;
  const int n0   = blockIdx.y * 64 + wave * 16;

  v8f acc = {};
  for (int kb = 0; kb < IN_Fc; kb += 4) {
    const int k0 = kb + hi * 2;
    v2f av, bv;
    av.x = h[(i0 + r) * IN_Fc + k0];
    av.y = h[(i0 + r) * IN_Fc + k0 + 1];
    bv.x = W[(k0    ) * HF + n0 + r];
    bv.y = W[(k0 + 1) * HF + n0 + r];
    acc = __builtin_amdgcn_wmma_f32_16x16x4_f32(false, av, false, bv,
                                                (short)0, acc, false, false);
  }
  for (int v = 0; v < 8; ++v)
    Wh[(i0 + v + hi * 8) * HF + n0 + r] = acc[v];
}

// ---------------------------------------------------------------------------
// Kernel 2: e_src[n,h] = dot(Wh[n,h,:], a_src[h,:]);  e_dst likewise.
// ---------------------------------------------------------------------------
__global__ __launch_bounds__(256) void k_attn_coef(const float* __restrict__ Wh,
                                                   const float* __restrict__ a_src,
                                                   const float* __restrict__ a_dst,
                                                   float* __restrict__ e_src,
                                                   float* __restrict__ e_dst) {
  const int idx = blockIdx.x * 256 + threadIdx.x;   // over N*H
  if (idx >= Nn * Hh) return;
  const int n = idx / Hh;
  const int hh = idx % Hh;
  const float* w = &Wh[n * HF + hh * Fc];
  float ss = 0.0f, sd = 0.0f;
  for (int f = 0; f < Fc; ++f) {
    const float v = w[f];
    ss += v * a_src[hh * Fc + f];
    sd += v * a_dst[hh * Fc + f];
  }
  e_src[idx] = ss;
  e_dst[idx] = sd;
}

// ---------------------------------------------------------------------------
// Kernel 3: per (row i, head h): rowmax and softmax denominator over j.
// Streams one adj row per block; 8-head running max/sum in registers, LDS tree
// reduction. Masked logits are exactly -1e9 so a fully-masked row yields the
// reference's uniform softmax (exp(0)=1 per element).
// ---------------------------------------------------------------------------
__global__ __launch_bounds__(256) void k_stats(const int* __restrict__ adj,
                                               const float* __restrict__ e_src,
                                               const float* __restrict__ e_dst,
                                               float* __restrict__ mrow,
                                               float* __restrict__ drow) {
  __shared__ float red[Hh * 256];
  const int i = blockIdx.x;
  const int tid = threadIdx.x;

  float es[Hh];
  for (int hh = 0; hh < Hh; ++hh) es[hh] = e_src[i * Hh + hh];

  float mx[Hh];
  for (int hh = 0; hh < Hh; ++hh) mx[hh] = NEG_INF;
  for (int j = tid; j < Nn; j += 256) {
    const int a = adj[i * Nn + j];
    for (int hh = 0; hh < Hh; ++hh) {
      const float lg = a ? lrelu(es[hh] + e_dst[j * Hh + hh]) : NEG_INF;
      mx[hh] = fmaxf(mx[hh], lg);
    }
  }
  for (int hh = 0; hh < Hh; ++hh) red[hh * 256 + tid] = mx[hh];
  __syncthreads();
  for (int s = 128; s > 0; s >>= 1) {
    if (tid < s)
      for (int hh = 0; hh < Hh; ++hh)
        red[hh * 256 + tid] = fmaxf(red[hh * 256 + tid], red[hh * 256 + tid + s]);
    __syncthreads();
  }
  float mh[Hh];
  for (int hh = 0; hh < Hh; ++hh) mh[hh] = red[hh * 256];
  __syncthreads();

  float sm[Hh];
  for (int hh = 0; hh < Hh; ++hh) sm[hh] = 0.0f;
  for (int j = tid; j < Nn; j += 256) {
    const int a = adj[i * Nn + j];
    for (int hh = 0; hh < Hh; ++hh) {
      const float lg = a ? lrelu(es[hh] + e_dst[j * Hh + hh]) : NEG_INF;
      sm[hh] += __expf(lg - mh[hh]);
    }
  }
  for (int hh = 0; hh < Hh; ++hh) red[hh * 256 + tid] = sm[hh];
  __syncthreads();
  for (int s = 128; s > 0; s >>= 1) {
    if (tid < s)
      for (int hh = 0; hh < Hh; ++hh)
        red[hh * 256 + tid] += red[hh * 256 + tid + s];
    __syncthreads();
  }
  if (tid < Hh) {
    mrow[i * Hh + tid] = mh[tid];
    drow[i * Hh + tid] = red[tid * 256];
  }
}

// ---------------------------------------------------------------------------
// Kernel 4: h'[i, h*64+f] = (1/denom[i,h]) * sum_j P[i,j,h] * Wh[j,h,f] + bias
// One block per (16-row i-tile, head); 4 waves each own a 16-wide f slab.
// Per 64-wide j-chunk:
//   - wave 0 issues a TDM tensor_load_to_lds of the 64x64 Wh tile (row pitch
//     512 floats) -> overlaps the DMA with the alpha-tile generation
//   - 128 threads compute P = exp(logit - rowmax) into a padded LDS tile
//   - wave 0 waits TENSORcnt==0 (per-wave counter), barrier releases all
//   - f32 WMMA 16x16x4 K-loop, A from Ps (stride-65, conflict-free columns),
//     B from the TDM-staged LDS tile (no global VMEM in the inner loop)
// ---------------------------------------------------------------------------
__global__ __launch_bounds__(128) void k_aggregate(const int* __restrict__ adj,
                                                   const float* __restrict__ Wh,
                                                   const float* __restrict__ e_src,
                                                   const float* __restrict__ e_dst,
                                                   const float* __restrict__ mrow,
                                                   const float* __restrict__ drow,
                                                   const float* __restrict__ bias,
                                                   float* __restrict__ out) {
  __shared__ float Whs[64 * 64];   // 16 KB: Whs[jj*64 + f], f relative to head
  __shared__ float Ps[16 * 65];    // padded alpha tile
  __shared__ float es_s[16];
  __shared__ float m_s[16];

  const int i0   = blockIdx.x * 16;
  const int hh   = blockIdx.y;
  const int tid  = threadIdx.x;
  const int lane = tid & 31;
  const int wave = tid >> 5;
  const int r    = lane & 15;
  const int hi   = lane >> 4;
  const int n0   = wave * 16;

  if (tid < 16) {
    es_s[tid] = e_src[(i0 + tid) * Hh + hh];
    m_s[tid]  = mrow[(i0 + tid) * Hh + hh];
  }

  const unsigned lds_off = (unsigned)(size_t)(&Whs[0]);  // LDS byte offset

  v8f acc = {};

  for (int jb = 0; jb < Nn; jb += 64) {
    __syncthreads();   // prior chunk's LDS consumers are done

    if (wave == 0) {
      // --- TDM descriptor: 64x64 f32 tile, row pitch 512 floats ---
      const unsigned long long ga =
          (unsigned long long)(size_t)(Wh + (size_t)jb * HF + hh * Fc);
      v4u g0;
      g0[0] = 1u;                                   // count=1, user mode
      g0[1] = lds_off;                              // lds_addr (bytes)
      g0[2] = (unsigned)(ga & 0xffffffffu);         // global_addr[31:0]
      g0[3] = (unsigned)((ga >> 32) & 0x1ffffffu)   // global_addr[56:32]
              | 0x80000000u;                        // type=2 ("image")
      v8i g1;
      g1[0] = 0x20000;        // workgroup_mask=0, data_size=2 (4B)
      g1[1] = 64 << 16;       // tensor_dim0 = 64
      g1[2] = 64 << 16;       // tensor_dim1 = 64
      g1[3] = 64 << 16;       // tile_dim0 = 64
      g1[4] = 64;             // tile_dim1 = 64, tile_dim2 = 0
      g1[5] = 512;            // tensor_dim0_stride = 512 (lo)
      g1[6] = 0;              // stride hi / dim1_stride lo
      g1[7] = 0;
      v4i g2 = {0, 0, 0, 0};  // 2-D transfer: higher dims unused
      v4i g3 = {0, 0, 0, 0};
      v8i g4 = {0, 0, 0, 0, 0, 0, 0, 0};
      __builtin_amdgcn_tensor_load_to_lds(g0, g1, g2, g3, g4, 0);
    }

    // Alpha tile generation overlaps the TDM transfer.
    for (int t = 0; t < 8; ++t) {
      const int row = t * 2 + (tid >> 6);
      const int jj  = tid & 63;
      const int j   = jb + jj;
      const int a   = adj[(i0 + row) * Nn + j];
      if (j + 64 < Nn)  // prefetch next chunk's adj (global_prefetch_b8)
        __builtin_prefetch(&adj[(i0 + row) * Nn + j + 64], 0, 0);
      const float lg = a ? lrelu(es_s[row] + e_dst[j * Hh + hh]) : NEG_INF;
      Ps[row * 65 + jj] = __expf(lg - m_s[row]);
    }

    if (wave == 0) __builtin_amdgcn_s_wait_tensorcnt(0);
    __syncthreads();

    for (int kb = 0; kb < 64; kb += 4) {
      const int k0 = kb + hi * 2;
      v2f av, bv;
      av.x = Ps[r * 65 + k0];
      av.y = Ps[r * 65 + k0 + 1];
      bv.x = Whs[(k0    ) * 64 + n0 + r];
      bv.y = Whs[(k0 + 1) * 64 + n0 + r];
      acc = __builtin_amdgcn_wmma_f32_16x16x4_f32(false, av, false, bv,
                                                  (short)0, acc, false, false);
    }
  }

  for (int v = 0; v < 8; ++v) {
    const int row = i0 + v + hi * 8;
    const int col = hh * Fc + n0 + r;
    out[row * HF + col] = acc[v] / drow[row * Hh + hh] + bias[col];
  }
}

// ---------------------------------------------------------------------------
// Launch. Inputs (setup_inputs order): h, adj, W, a_src, a_dst, bias.
// Workspace layout (floats): Wh [N*HF] | e_src [N*H] | e_dst [N*H] |
//                            mrow [N*H] | drow [N*H]   (~8.5 MB total)
// ---------------------------------------------------------------------------
extern "C" void kernel_launch(void* const* d_in, const int* in_sizes, int n_in,
                              void* d_out, int out_size, void* d_ws, size_t ws_size,
                              hipStream_t stream) {
  (void)in_sizes; (void)n_in; (void)out_size; (void)ws_size;
  const float* h     = (const float*)d_in[0];
  const int*   adj   = (const int*)  d_in[1];
  const float* W     = (const float*)d_in[2];
  const float* a_src = (const float*)d_in[3];
  const float* a_dst = (const float*)d_in[4];
  const float* bias  = (const float*)d_in[5];
  float* out = (float*)d_out;

  float* ws    = (float*)d_ws;
  float* Wh    = ws;                       // N*HF
  float* e_src = Wh    + (size_t)Nn * HF;  // N*H
  float* e_dst = e_src + (size_t)Nn * Hh;
  float* mrow  = e_dst + (size_t)Nn * Hh;
  float* drow  = mrow  + (size_t)Nn * Hh;

  k_wh<<<dim3(Nn / 16, HF / 64), 128, 0, stream>>>(h, W, Wh);
  k_attn_coef<<<(Nn * Hh) / 256, 256, 0, stream>>>(Wh, a_src, a_dst, e_src, e_dst);
  k_stats<<<Nn, 256, 0, stream>>>(adj, e_src, e_dst, mrow, drow);
  k_aggregate<<<dim3(Nn / 16, Hh), 128, 0, stream>>>(adj, Wh, e_src, e_dst,
                                                     mrow, drow, bias, out);
}